// PolyIoULoss_1202590843407
// MI455X (gfx1250) — compile-verified
//
#include <hip/hip_runtime.h>
#include <hip/hip_bf16.h>
#include <math.h>

typedef float v2f __attribute__((ext_vector_type(2)));
typedef float v8f __attribute__((ext_vector_type(8)));

#define TPB 128
#define STRIDE 25  // padded per-thread LDS stride: gcd(25,64)=1 -> conflict-free

__global__ __launch_bounds__(TPB) void poly_iou_kernel(
    const float* __restrict__ pred, const float* __restrict__ target,
    int n, float inv_n, float* __restrict__ partials)
{
    __shared__ float sx[TPB * STRIDE];
    __shared__ float sy[TPB * STRIDE];
    __shared__ float sa[TPB * STRIDE];
    __shared__ float wsum[TPB / 32];

    const int tid = threadIdx.x;
    const int i   = blockIdx.x * TPB + tid;

    float* mx = &sx[tid * STRIDE];
    float* my = &sy[tid * STRIDE];
    float* ma = &sa[tid * STRIDE];

    float lsum = 0.0f;

    if (i < n) {
        // ---- load two quads (2 x b128 each, fully coalesced) ----
        const float4* p4 = reinterpret_cast<const float4*>(pred)   + (size_t)i * 2;
        const float4* t4 = reinterpret_cast<const float4*>(target) + (size_t)i * 2;
        float4 pa = p4[0], pb = p4[1];
        float4 ta = t4[0], tb = t4[1];
        float ax[4] = {pa.x, pa.z, pb.x, pb.z};
        float ay[4] = {pa.y, pa.w, pb.y, pb.w};
        float bx[4] = {ta.x, ta.z, tb.x, tb.z};
        float by[4] = {ta.y, ta.w, tb.y, tb.w};

        // ---- shoelace areas ----
        float s1 = 0.f, s2 = 0.f;
#pragma unroll
        for (int k = 0; k < 4; ++k) {
            int km = (k + 3) & 3;
            s1 += ax[k] * ay[km] - ax[km] * ay[k];
            s2 += bx[k] * by[km] - bx[km] * by[k];
        }
        float area1 = 0.5f * fabsf(s1);
        float area2 = 0.5f * fabsf(s2);

        // ---- 16 edge-pair intersections + containment partial sums ----
        unsigned mk = 0u;
        float st1[4] = {0.f, 0.f, 0.f, 0.f};
        float st2[4] = {0.f, 0.f, 0.f, 0.f};
#pragma unroll
        for (int a = 0; a < 4; ++a) {
            float x1 = ax[a], y1 = ay[a];
            float x2 = ax[(a + 1) & 3], y2 = ay[(a + 1) & 3];
#pragma unroll
            for (int b = 0; b < 4; ++b) {
                float x3 = bx[b], y3 = by[b];
                float x4 = bx[(b + 1) & 3], y4 = by[(b + 1) & 3];
                float num  = (x1 - x2) * (y3 - y4) - (y1 - y2) * (x3 - x4);
                float dent = (x1 - x3) * (y3 - y4) - (y1 - y3) * (x3 - x4);
                float denu = (x2 - x1) * (y1 - y3) - (y2 - y1) * (x1 - x3);
                float tng = dent / num;             // stop-grad path: NaN/inf -> mask false
                float ung = denu / num;
                bool m = (tng > 0.f) && (tng < 1.f) && (ung > 0.f) && (ung < 1.f);
                float t = dent / (num + 1e-6f);
                int idx = a * 4 + b;
                mx[idx] = x1 + t * (x2 - x1);
                my[idx] = y1 + t * (y2 - y1);
                if (m) mk |= (1u << idx);
                st1[a] += 0.5f * fabsf((x3 - x1) * (y4 - y1) - (y3 - y1) * (x4 - x1));
                st2[b] += 0.5f * fabsf((x1 - x3) * (y2 - y3) - (x2 - x3) * (y1 - y3));
            }
        }
#pragma unroll
        for (int a = 0; a < 4; ++a) {
            mx[16 + a] = ax[a]; my[16 + a] = ay[a];
            if (fabsf(st1[a] - area2) < 0.001f * area2) mk |= (1u << (16 + a));
            mx[20 + a] = bx[a]; my[20 + a] = by[a];
            if (fabsf(st2[a] - area1) < 0.001f * area1) mk |= (1u << (20 + a));
        }

        // ---- masked centroid ----
        float cnt = 0.f, cx = 0.f, cy = 0.f;
        unsigned rem = mk;
        while (rem) {
            int j = __ffs(rem) - 1; rem &= rem - 1;
            cnt += 1.f; cx += mx[j]; cy += my[j];
        }
        float dnm = fmaxf(cnt, 1.f);
        cx /= dnm; cy /= dnm;

        // ---- angles of masked points ----
        rem = mk;
        while (rem) {
            int j = __ffs(rem) - 1; rem &= rem - 1;
            ma[j] = atan2f(my[j] - cy, mx[j] - cx);
        }

        // ---- shoelace over angularly-sorted masked points via cyclic successors.
        //      Tie-break matches stable argsort: among equal angles, next-larger
        //      index is successor; wrap adds 2*pi. m<=2 naturally gives area 0. ----
        const float TWO_PI = 6.2831853071795864f;
        float s = 0.f;
        rem = mk;
        while (rem) {
            int j = __ffs(rem) - 1; rem &= rem - 1;
            float aj = ma[j];
            float bd = 1e30f; int bi = -1;
            unsigned rem2 = mk & ~(1u << j);
            while (rem2) {
                int k = __ffs(rem2) - 1; rem2 &= rem2 - 1;
                float d = ma[k] - aj;
                if (d < 0.f || (d == 0.f && k < j)) d += TWO_PI;
                if (d < bd) { bd = d; bi = k; }   // strict < keeps smallest k on ties
            }
            if (bi >= 0) s += mx[j] * my[bi] - my[j] * mx[bi];
        }
        float overlap = 0.5f * fabsf(s);

        float iou = overlap / (area1 + area2 - overlap + 1e-6f);
        iou = fminf(fmaxf(iou, 1e-6f), 1.0f);
        lsum = (1.0f - iou) * inv_n;
    }

    // ---- wave32 reduction: exact f32 WMMA folds lane L with lane L+16.
    //      A = ones(16x4), B = {lsum, 0} -> D[m][n] = lsum[n] + lsum[n+16].
    //      EXEC is all-ones here (no early returns above). ----
    v2f A = {1.0f, 1.0f};
    v2f B = {lsum, 0.0f};
    v8f C = {0.f, 0.f, 0.f, 0.f, 0.f, 0.f, 0.f, 0.f};
    C = __builtin_amdgcn_wmma_f32_16x16x4_f32(false, A, false, B, (short)0, C,
                                              false, false);
    float r = C[0];                 // every lane: lsum[L&15] + lsum[(L&15)+16]
    r += __shfl_xor(r, 1, 32);
    r += __shfl_xor(r, 2, 32);
    r += __shfl_xor(r, 4, 32);
    r += __shfl_xor(r, 8, 32);      // full wave sum in all lanes

    if ((tid & 31) == 0) wsum[tid >> 5] = r;
    __syncthreads();
    if (tid == 0) {
        float bs = 0.f;
#pragma unroll
        for (int w = 0; w < TPB / 32; ++w) bs += wsum[w];
        partials[blockIdx.x] = bs;
    }
}

// Deterministic fixed-order final reduction (no float atomics -> graph replays
// reproduce the value bit-exactly).
__global__ __launch_bounds__(256) void final_reduce(
    const float* __restrict__ partials, int nblocks, float* __restrict__ out)
{
    __shared__ float sm[256];
    float s = 0.f;
    for (int t = threadIdx.x; t < nblocks; t += 256) s += partials[t];
    sm[threadIdx.x] = s;
    __syncthreads();
    for (int off = 128; off > 0; off >>= 1) {
        if ((int)threadIdx.x < off) sm[threadIdx.x] += sm[threadIdx.x + off];
        __syncthreads();
    }
    if (threadIdx.x == 0) out[0] = sm[0];
}

extern "C" void kernel_launch(void* const* d_in, const int* in_sizes, int n_in,
                              void* d_out, int out_size, void* d_ws, size_t ws_size,
                              hipStream_t stream) {
    const float* pred   = (const float*)d_in[0];
    const float* target = (const float*)d_in[1];
    int n = in_sizes[0] / 8;                  // N polygon pairs
    float* partials = (float*)d_ws;           // nblocks floats of scratch
    int blocks = (n + TPB - 1) / TPB;
    poly_iou_kernel<<<blocks, TPB, 0, stream>>>(pred, target, n,
                                                1.0f / (float)n, partials);
    final_reduce<<<1, 256, 0, stream>>>(partials, blocks, (float*)d_out);
}